// _ProposalLayer_28930899706155
// MI455X (gfx1250) — compile-verified
//
#include <hip/hip_runtime.h>
#include <cstdint>

// ---------------- problem constants (from reference) ----------------
#define PB       32          // batch
#define PH       64          // feat H
#define PW       64          // feat W
#define PA       9           // anchors
#define TK_K     300         // post_nms_topN
#define CAND     512         // candidate buffer for sort (>= K + dup slack)
#define OUTC     66          // 1 + 64 + 1

typedef __attribute__((ext_vector_type(2))) float v2f;
typedef __attribute__((ext_vector_type(8))) float v8f;

#if __has_builtin(__builtin_amdgcn_wmma_f32_16x16x4_f32)
#define USE_WMMA_COMBINE 1
#else
#define USE_WMMA_COMBINE 0
#endif

// per-anchor width/height (x2-x1+1, y2-y1+1) -- derived with numpy banker's rounding
__device__ __constant__ float c_AW[PA] = {92.f,184.f,368.f,64.f,128.f,256.f,44.f,88.f,176.f};
__device__ __constant__ float c_AH[PA] = {48.f,96.f,192.f,64.f,128.f,256.f,88.f,176.f,352.f};

// monotonic float->uint key: descending float == descending unsigned key
__device__ __forceinline__ unsigned f2key(float f) {
    unsigned fb = __float_as_uint(f);
    return fb ^ ((fb & 0x80000000u) ? 0xFFFFFFFFu : 0x80000000u);
}
__device__ __forceinline__ float key2f(unsigned k) {
    unsigned fb = (k & 0x80000000u) ? (k ^ 0x80000000u) : ~k;
    return __uint_as_float(fb);
}

// CDNA5 async data mover: each of 1024 lanes copies 16B global -> LDS
// (GLOBAL_LOAD_ASYNC_TO_LDS_B128, ASYNCcnt-tracked; CDNA5 ISA)
__device__ __forceinline__ void issue_plane_async(const float* __restrict__ gsrc,
                                                  float* lds_dst, int tid) {
    unsigned ldsoff = (unsigned)(uintptr_t)lds_dst + (unsigned)(tid * 16);
    const char* gp  = (const char*)gsrc + tid * 16;
    asm volatile("global_load_async_to_lds_b128 %0, %1, off"
                 :: "v"(ldsoff), "v"(gp) : "memory");
}
__device__ __forceinline__ void wait_async_le1() {
    asm volatile("s_wait_asynccnt 0x1" ::: "memory");
}
__device__ __forceinline__ void wait_async_le0() {
    asm volatile("s_wait_asynccnt 0x0" ::: "memory");
}

// ---------------- kernel 1: exact top-300 per batch (radix select) ----------------
__global__ __launch_bounds__(1024)
void topk_kernel(const float* __restrict__ scores_in,
                 int* __restrict__ top_idx, float* __restrict__ top_score) {
    __shared__ unsigned hist[2048];
    __shared__ float    plane[2][PH*PW];        // double-buffered 64x64 planes (32 KB)
    __shared__ unsigned s_bkt, s_rem;
    __shared__ unsigned cand_key[CAND];
    __shared__ int      cand_idx[CAND];
    __shared__ int      s_ncand;

    const int b   = blockIdx.x;
    const int tid = threadIdx.x;
    const int nth = 1024;
    // objectness scores live in channels [9,18): plane base for anchor a
    const float* sbase = scores_in + (((size_t)b * (2*PA) + PA) << 12);

    unsigned rem   = TK_K;   // rank still to locate
    unsigned tpref = 0;      // accumulated high bits of the threshold key
    const int shifts[3] = {21, 10, 0};
    const int nbits [3] = {11, 11, 10};

    for (int p = 0; p < 3; ++p) {
        const int nb   = 1 << nbits[p];
        const int sh   = shifts[p];
        const int hish = sh + nbits[p];
        const unsigned mask = (unsigned)(nb - 1);
        for (int i = tid; i < 2048; i += nth) hist[i] = 0u;
        issue_plane_async(sbase, plane[0], tid);               // prefetch plane 0
        for (int a = 0; a < PA; ++a) {
            const int buf = a & 1;
            if (a + 1 < PA) {                                  // overlap DMA of a+1
                issue_plane_async(sbase + (size_t)(a+1) * (PH*PW), plane[(a+1)&1], tid);
                wait_async_le1();
            } else {
                wait_async_le0();
            }
            __syncthreads();
            for (int e = tid; e < PH*PW; e += nth) {
                unsigned key = f2key(plane[buf][e]);
                bool ok = (p == 0) || ((key >> hish) == tpref);
                if (ok) atomicAdd(&hist[(key >> sh) & mask], 1u);
            }
            __syncthreads();   // plane consumed before it can be restaged
        }
        if (tid == 0) {
            unsigned cum = 0, sel = 0, nr = 1;
            for (int bkt = nb - 1; bkt >= 0; --bkt) {
                unsigned c = hist[bkt];
                if (cum + c >= rem) { sel = (unsigned)bkt; nr = rem - cum; break; }
                cum += c;
            }
            s_bkt = sel; s_rem = nr;
        }
        __syncthreads();
        tpref = (tpref << nbits[p]) | s_bkt;
        rem   = s_rem;
        __syncthreads();
    }
    const unsigned T = tpref;   // exact 300th-largest key (11+11+10 = 32 bits)

    // gather all candidates with key >= T (count in [300, 300+dup slack])
    for (int i = tid; i < CAND; i += nth) { cand_key[i] = 0u; cand_idx[i] = 0x7FFFFFFF; }
    if (tid == 0) s_ncand = 0;
    issue_plane_async(sbase, plane[0], tid);
    __syncthreads();
    for (int a = 0; a < PA; ++a) {
        const int buf = a & 1;
        if (a + 1 < PA) {
            issue_plane_async(sbase + (size_t)(a+1) * (PH*PW), plane[(a+1)&1], tid);
            wait_async_le1();
        } else {
            wait_async_le0();
        }
        __syncthreads();
        for (int e = tid; e < PH*PW; e += nth) {
            unsigned key = f2key(plane[buf][e]);
            if (key >= T) {
                int pos = atomicAdd(&s_ncand, 1);
                if (pos < CAND) { cand_key[pos] = key; cand_idx[pos] = e * PA + a; }
            }
        }
        __syncthreads();
    }

    // bitonic sort CAND entries: key descending, idx ascending (matches lax.top_k order)
    for (int k2 = 2; k2 <= CAND; k2 <<= 1) {
        for (int j = k2 >> 1; j > 0; j >>= 1) {
            for (int t = tid; t < CAND; t += nth) {
                int ixj = t ^ j;
                if (ixj > t) {
                    unsigned ka = cand_key[t], kb = cand_key[ixj];
                    int ia = cand_idx[t], ib = cand_idx[ixj];
                    bool before = (ka > kb) || (ka == kb && ia < ib);
                    bool up = ((t & k2) == 0);
                    if (up ? !before : before) {
                        cand_key[t] = kb; cand_key[ixj] = ka;
                        cand_idx[t] = ib; cand_idx[ixj] = ia;
                    }
                }
            }
            __syncthreads();
        }
    }
    if (tid < TK_K) {
        top_idx  [b * TK_K + tid] = cand_idx[tid];
        top_score[b * TK_K + tid] = key2f(cand_key[tid]);
    }
}

// ---------------- kernel 2: transform only the 300 winners per batch ----------------
// One wave per output row (64 coords = 16 time-groups x 4).
// The coordinate combine [x1,y1,x2,y2] = [pcx,pcy,pw,ph] . M is a true 16x16x4 matmul:
//   A (16x4): row g = group g's [pcx,pcy,pw,ph]; ISA layout puts K=0,1 in lanes 0-15
//   and K=2,3 in lanes 16-31 -> lower half-lanes produce pcx,pcy; upper produce pw,ph.
//   B (4x16): constant coefficients, cols 0..3 = x1/y1/x2/y2, cols 4..15 = 0.
//   D (16x16): lane n<4 of each half holds row m's coord n in acc[m%8].
__global__ __launch_bounds__(256)
void proposal_kernel(const float* __restrict__ bbox, const float* __restrict__ im_info,
                     const int* __restrict__ top_idx, const float* __restrict__ top_score,
                     float* __restrict__ out) {
    const int lane = threadIdx.x & 31;
    const int wave = threadIdx.x >> 5;
    const int row  = blockIdx.x * 8 + wave;     // b*K + r  (grid = B*K/8 blocks)
    const int b    = row / TK_K;

    const int idx = top_idx[row];
    const int a   = idx % PA;
    const int pix = idx / PA;                   // h*64 + w
    const int wp  = pix & 63, hp = pix >> 6;

    const float w  = c_AW[a], h = c_AH[a];
    const float cx = 8.0f + 16.0f * (float)wp;  // all base anchors centered at 7.5
    const float cy = 8.0f + 16.0f * (float)hp;

    const int  g  = lane & 15;                  // time group this lane feeds
    const bool hi = lane >= 16;                 // upper half-lanes: K=2,3 (pw,ph)
    const int  c0 = g * 4 + (hi ? 2 : 0);       // channels d0,d1 (lo) / d2,d3 (hi)
    const float* p = bbox + (((size_t)(b * (PA*64) + a * 64 + c0)) << 12) + pix;
    const float da = p[0];
    const float db = p[4096];

    v2f A;
    if (!hi) { A.x = da * w + cx;   A.y = db * h + cy;  }   // pcx, pcy
    else     { A.x = expf(da) * w;  A.y = expf(db) * h; }   // pw, ph

    const float xmax = im_info[b*3 + 1] - 1.0f;
    const float ymax = im_info[b*3 + 0] - 1.0f;
    float* o = out + (size_t)row * OUTC;

#if USE_WMMA_COMBINE
    // B (4x16) in ISA layout: VGPR0 = rows K=0 (lanes 0-15) / K=2 (lanes 16-31),
    //                         VGPR1 = rows K=1 / K=3.  Coefficient matrix M:
    //   M[0][*]=coeff(pcx)={1,0,1,0,..}; M[1]=coeff(pcy)={0,1,0,1,..};
    //   M[2]=coeff(pw)={-.5,0,.5,0,..};  M[3]=coeff(ph)={0,-.5,0,.5,..}
    v2f Bv; Bv.x = 0.0f; Bv.y = 0.0f;
    if (!hi) {
        if (g == 0 || g == 2) Bv.x = 1.0f;
        if (g == 1 || g == 3) Bv.y = 1.0f;
    } else {
        if (g == 0) Bv.x = -0.5f; else if (g == 2) Bv.x = 0.5f;
        if (g == 1) Bv.y = -0.5f; else if (g == 3) Bv.y = 0.5f;
    }
    v8f acc = {};
    acc = __builtin_amdgcn_wmma_f32_16x16x4_f32(
        /*neg_a=*/false, A, /*neg_b=*/false, Bv,
        /*c_mod=*/(short)0, acc, /*reuse_a=*/false, /*reuse_b=*/false);
    // D layout: acc[r] = D[m][n], m = r + (hi?8:0), n = g; only n<4 meaningful
    if (g < 4) {
        const float cmax  = (g & 1) ? ymax : xmax;
        const int   mbase = hi ? 8 : 0;
        #pragma unroll
        for (int r = 0; r < 8; ++r) {
            float v = fminf(fmaxf(acc[r], 0.0f), cmax);
            o[1 + (mbase + r) * 4 + g] = v;
        }
    }
#else
    // scalar fallback: exchange halves so low lanes see pw,ph
    float ox = __shfl_xor(A.x, 16);
    float oy = __shfl_xor(A.y, 16);
    if (!hi) {
        float pcx = A.x, pcy = A.y, pw = ox, ph = oy;
        float x1 = fminf(fmaxf(pcx - 0.5f*pw, 0.0f), xmax);
        float y1 = fminf(fmaxf(pcy - 0.5f*ph, 0.0f), ymax);
        float x2 = fminf(fmaxf(pcx + 0.5f*pw, 0.0f), xmax);
        float y2 = fminf(fmaxf(pcy + 0.5f*ph, 0.0f), ymax);
        o[1 + g*4 + 0] = x1; o[1 + g*4 + 1] = y1;
        o[1 + g*4 + 2] = x2; o[1 + g*4 + 3] = y2;
    }
#endif
    if (lane == 0) { o[0] = (float)b; o[OUTC - 1] = top_score[row]; }
}

extern "C" void kernel_launch(void* const* d_in, const int* in_sizes, int n_in,
                              void* d_out, int out_size, void* d_ws, size_t ws_size,
                              hipStream_t stream) {
    const float* scores = (const float*)d_in[0];  // (32, 18, 64, 64)
    const float* bbox   = (const float*)d_in[1];  // (32, 576, 64, 64)
    const float* iminfo = (const float*)d_in[2];  // (32, 3)
    (void)in_sizes; (void)n_in; (void)out_size; (void)ws_size;

    int*   top_idx   = (int*)d_ws;
    float* top_score = (float*)((char*)d_ws + (size_t)PB * TK_K * sizeof(int));

    topk_kernel<<<PB, 1024, 0, stream>>>(scores, top_idx, top_score);
    proposal_kernel<<<(PB * TK_K) / 8, 256, 0, stream>>>(
        bbox, iminfo, top_idx, top_score, (float*)d_out);
}